// BatchATSSAssigner_55989193671319
// MI455X (gfx1250) — compile-verified
//
#include <hip/hip_runtime.h>
#include <hip/hip_bf16.h>
#include <float.h>

#define B_   32
#define G_   32
#define P_   8400
#define BG_  (B_ * G_)
#define NC_  80
#define EPS_ 1e-9f

typedef float v2f __attribute__((ext_vector_type(2)));
typedef float v8f __attribute__((ext_vector_type(8)));

// ---------------------------------------------------------------------------
// Kernel 1: pairwise squared center distances via V_WMMA_F32_16X16X4_F32.
// d^2(g,p) = [gx, gy, |gc|^2, 1] . [-2px, -2py, 1, |pc|^2]   (exact rank-4 GEMM)
// ---------------------------------------------------------------------------
__global__ __launch_bounds__(32) void dist2_wmma(
    const float4* __restrict__ gt,    // [BG] boxes
    const float4* __restrict__ pri,   // [P]  boxes
    float* __restrict__ dist2)        // [BG][P]
{
    const int lane = threadIdx.x;
    const int hi   = lane >> 4;       // 0: K0/K1 half, 1: K2/K3 half
    const int l15  = lane & 15;
    const int row0 = blockIdx.y * 16; // over BG (1024 = 64*16)
    const int col0 = blockIdx.x * 16; // over P  (8400 = 525*16)

    const float4 g  = gt[row0 + l15];
    const float gx = 0.5f * (g.x + g.z);
    const float gy = 0.5f * (g.y + g.w);
    const float sg = gx * gx + gy * gy;

    v2f a;
    a.x = hi ? sg   : gx;
    a.y = hi ? 1.0f : gy;

    const float4 q  = pri[col0 + l15];
    const float px = 0.5f * (q.x + q.z);
    const float py = 0.5f * (q.y + q.w);
    const float sp = px * px + py * py;

    v2f bm;
    bm.x = hi ? 1.0f : -2.0f * px;
    bm.y = hi ? sp   : -2.0f * py;

    v8f c = {0.f, 0.f, 0.f, 0.f, 0.f, 0.f, 0.f, 0.f};
    c = __builtin_amdgcn_wmma_f32_16x16x4_f32(
            /*neg_a=*/false, a, /*neg_b=*/false, bm,
            /*c_mod=*/(short)0, c, /*reuse_a=*/false, /*reuse_b=*/false);

    const int m0 = row0 + hi * 8;
#pragma unroll
    for (int r = 0; r < 8; ++r)
        dist2[(long)(m0 + r) * P_ + col0 + l15] = c[r];
}

// ---------------------------------------------------------------------------
// Kernel 2: one wave32 per (b,g). The dist^2 row is streamed through LDS with
// double-buffered GLOBAL_LOAD_ASYNC_TO_LDS_B128 (one wave-wide async op stages
// a 512B chunk; s_wait_asynccnt overlaps DMA with top-9 insertion work).
// Then ATSS threshold over the 27 candidates, center-in-gt test, and the row
// is rewritten in place as the sparse pos_mask.
// ---------------------------------------------------------------------------
__global__ __launch_bounds__(32) void atss_select(
    float* dist2,                       // [BG][P]; read as dist^2, rewritten as pos_mask
    const float4* __restrict__ gt,      // [BG]
    const float4* __restrict__ pri,     // [P]
    const float*  __restrict__ padFlag) // [BG]
{
    const int bg   = blockIdx.x;
    const int lane = threadIdx.x;

    __shared__ __align__(16) float sBuf[2][128];   // async staging, 2 x 512B
    __shared__ float sD[32 * 9];
    __shared__ int   sI[32 * 9];
    __shared__ int   candIdx[27];
    __shared__ float candO[27];
    __shared__ int   candPos[27];
    __shared__ float sThr;

    const float4 g    = gt[bg];
    const bool  valid = padFlag[bg] > 0.0f;
    const int starts[3] = {0, 6400, 8000};
    const int ns[3]     = {6400, 1600, 400};
    float* row = dist2 + (long)bg * P_;

    // LDS aperture holds the byte offset in addr[31:0]; truncate flat address.
    const unsigned ldsBase =
        (unsigned)(unsigned long long)(uintptr_t)(&sBuf[0][0]) + (unsigned)lane * 16u;

    for (int lvl = 0; lvl < 3; ++lvl) {
        float d9[9]; int i9[9];
#pragma unroll
        for (int j = 0; j < 9; ++j) { d9[j] = FLT_MAX; i9[j] = 0; }

        const int n = ns[lvl], st = starts[lvl];
        const int nChunks = (n + 127) / 128;       // 50 / 13 / 4 chunks of 128 floats

        // prologue: stage chunk 0
        {
            const unsigned gof = (unsigned)(st * 4) + (unsigned)lane * 16u;
            asm volatile("global_load_async_to_lds_b128 %0, %1, %2"
                         :: "v"(ldsBase), "v"(gof), "s"(row) : "memory");
        }
        for (int c = 0; c < nChunks; ++c) {
            if (c + 1 < nChunks) {                 // stage next chunk into other buffer
                const unsigned lds = ldsBase + (unsigned)(((c + 1) & 1) * 512);
                const unsigned gof = (unsigned)((st + (c + 1) * 128) * 4) + (unsigned)lane * 16u;
                asm volatile("global_load_async_to_lds_b128 %0, %1, %2"
                             :: "v"(lds), "v"(gof), "s"(row) : "memory");
                asm volatile("s_wait_asynccnt 0x1" ::: "memory");
            } else {
                asm volatile("s_wait_asynccnt 0x0" ::: "memory");
            }
            const float4* sB4 = (const float4*)(&sBuf[c & 1][0]);
            const float4 vv = sB4[lane];
            const int base = c * 128 + lane * 4;
            const float vk[4] = {vv.x, vv.y, vv.z, vv.w};
#pragma unroll
            for (int k = 0; k < 4; ++k) {
                const int i = base + k;
                if (i < n) {
                    const float v = vk[k];
#pragma unroll
                    for (int j = 8; j >= 0; --j) {  // sorted-ascending insertion
                        if (v < d9[j]) {
                            if (j < 8) { d9[j + 1] = d9[j]; i9[j + 1] = i9[j]; }
                            d9[j] = v; i9[j] = i;
                        }
                    }
                }
            }
        }
#pragma unroll
        for (int j = 0; j < 9; ++j) { sD[lane * 9 + j] = d9[j]; sI[lane * 9 + j] = i9[j]; }
        __syncthreads();
        if (lane == 0) {                          // 9-way selection over 288 entries
            for (int s = 0; s < 9; ++s) {
                float best = FLT_MAX; int bi = 0;
                for (int e = 0; e < 288; ++e) {
                    const float dv = sD[e];
                    if (dv < best) { best = dv; bi = e; }
                }
                candIdx[lvl * 9 + s] = st + sI[bi];
                sD[bi] = FLT_MAX;
            }
        }
        __syncthreads();
    }

    // candidate overlaps (recomputed IoU; union clamped at 1e-6 like reference)
    if (lane < 27) {
        const float4 q = pri[candIdx[lane]];
        const float lx = fmaxf(g.x, q.x), ly = fmaxf(g.y, q.y);
        const float rx = fminf(g.z, q.z), ry = fminf(g.w, q.w);
        const float inter = fmaxf(rx - lx, 0.f) * fmaxf(ry - ly, 0.f);
        const float ag = (g.z - g.x) * (g.w - g.y);
        const float ap = (q.z - q.x) * (q.w - q.y);
        const float o  = inter / fmaxf(ag + ap - inter, 1e-6f);
        candO[lane] = valid ? o : 0.0f;           // is_in_candidate==0 for padded gts
    }
    __syncthreads();
    if (lane == 0) {
        float s = 0.f, ss = 0.f;
        for (int c = 0; c < 27; ++c) { const float v = candO[c]; s += v; ss += v * v; }
        const float mean = s / 27.0f;
        const float var  = (ss - 27.0f * mean * mean) / 26.0f;  // ddof=1
        sThr = mean + sqrtf(fmaxf(var, 0.0f));
    }
    __syncthreads();
    if (lane < 27) {
        const float4 q = pri[candIdx[lane]];
        const float px = 0.5f * (q.x + q.z), py = 0.5f * (q.y + q.w);
        const float dmin = fminf(fminf(px - g.x, py - g.y), fminf(g.z - px, g.w - py));
        const bool inGt = dmin > EPS_;
        const bool pos  = (candO[lane] > sThr) && inGt && valid;
        candPos[lane] = pos ? 1 : 0;
    }
    __syncthreads();

    // rewrite the row as pos_mask (all dist^2 reads for this row are done)
    for (int i = lane; i < P_; i += 32) {
        float v = 0.0f;
#pragma unroll
        for (int c = 0; c < 27; ++c)
            if (candPos[c] && candIdx[c] == i) v = 1.0f;
        row[i] = v;
    }
}

// ---------------------------------------------------------------------------
// Kernel 3: one thread per (b,p). G=32 == wave width => gt mask in a uint32.
// Coalesced column reads of pos_mask; overlaps recomputed (cheaper than HBM).
// ---------------------------------------------------------------------------
__global__ __launch_bounds__(256) void atss_assign(
    const float* __restrict__ posMask,   // [BG][P]
    const float4* __restrict__ gt,       // [BG]
    const float4* __restrict__ pri,      // [P]
    const int*   __restrict__ gtLabels,  // [BG]
    const float4* __restrict__ pred,     // [B][P]
    float* __restrict__ outLabels,       // [B*P]
    float* __restrict__ outBoxes,        // [B*P][4]
    float* __restrict__ outFg,           // [B*P]
    float* __restrict__ wsLabel,         // [B*P] scratch
    float* __restrict__ wsIou)           // [B*P] scratch
{
    const int t = blockIdx.x * blockDim.x + threadIdx.x;
    if (t >= B_ * P_) return;
    const int b = t / P_, p = t % P_;

    const float4 q  = pri[p];
    const float ap  = (q.z - q.x) * (q.w - q.y);

    unsigned mask = 0u;
    float fg = 0.0f, bestOv = -FLT_MAX;
    int bestG = 0;
    for (int gi = 0; gi < G_; ++gi) {
        const long  idx = (long)(b * G_ + gi) * P_ + p;
        const float pm  = posMask[idx];
        const float4 gb = gt[b * G_ + gi];
        const float lx = fmaxf(gb.x, q.x), ly = fmaxf(gb.y, q.y);
        const float rx = fminf(gb.z, q.z), ry = fminf(gb.w, q.w);
        const float inter = fmaxf(rx - lx, 0.f) * fmaxf(ry - ly, 0.f);
        const float ag = (gb.z - gb.x) * (gb.w - gb.y);
        const float ov = inter / fmaxf(ag + ap - inter, 1e-6f);
        if (ov > bestOv) { bestOv = ov; bestG = gi; }   // first max (argmax semantics)
        if (pm > 0.5f)   { mask |= 1u << gi; fg += 1.0f; }
    }
    if (fg > 1.0f) mask = 1u << bestG;                   // multi-assignment fix

    const int  gtIdx = mask ? (__ffs(mask) - 1) : 0;     // argmax over {0,1}
    const bool fgm   = mask != 0u;
    const int  labI  = fgm ? gtLabels[b * G_ + gtIdx] : NC_;

    // masked max IoU(gt, pred) over assigned gts
    const float4 pr  = pred[(long)b * P_ + p];
    const float apd  = fmaxf(pr.z - pr.x, 0.f) * fmaxf(pr.w - pr.y, 0.f);
    float maxIou = 0.0f;
    unsigned m = mask;
    while (m) {
        const int gi = __ffs(m) - 1; m &= m - 1;
        const float4 gb = gt[b * G_ + gi];
        const float lx = fmaxf(gb.x, pr.x), ly = fmaxf(gb.y, pr.y);
        const float rx = fminf(gb.z, pr.z), ry = fminf(gb.w, pr.w);
        const float inter = fmaxf(rx - lx, 0.f) * fmaxf(ry - ly, 0.f);
        const float ag = fmaxf(gb.z - gb.x, 0.f) * fmaxf(gb.w - gb.y, 0.f);
        maxIou = fmaxf(maxIou, inter / (ag + apd - inter + EPS_));
    }

    outLabels[t] = (float)labI;
    ((float4*)outBoxes)[t] = gt[b * G_ + gtIdx];
    outFg[t]   = fgm ? 1.0f : 0.0f;
    wsLabel[t] = (float)labI;
    wsIou[t]   = maxIou;
}

// ---------------------------------------------------------------------------
// Kernel 4: scores = one_hot(label)[:80] * iou. One thread per 4 classes ->
// one coalesced global_store_b128 each (this 86 MB tensor dominates traffic).
// ---------------------------------------------------------------------------
__global__ __launch_bounds__(256) void atss_scores(
    const float* __restrict__ wsLabel,
    const float* __restrict__ wsIou,
    float4* __restrict__ outScores)      // [B*P][20] float4
{
    const long t = (long)blockIdx.x * blockDim.x + threadIdx.x;
    if (t >= (long)B_ * P_ * (NC_ / 4)) return;
    const long bp = t / (NC_ / 4);
    const int  c0 = (int)(t % (NC_ / 4)) * 4;
    const float lab = wsLabel[bp];
    const float iou = wsIou[bp];
    float4 r;
    r.x = ((float)(c0 + 0) == lab) ? iou : 0.0f;
    r.y = ((float)(c0 + 1) == lab) ? iou : 0.0f;
    r.z = ((float)(c0 + 2) == lab) ? iou : 0.0f;
    r.w = ((float)(c0 + 3) == lab) ? iou : 0.0f;
    outScores[t] = r;
}

extern "C" void kernel_launch(void* const* d_in, const int* in_sizes, int n_in,
                              void* d_out, int out_size, void* d_ws, size_t ws_size,
                              hipStream_t stream) {
    (void)in_sizes; (void)n_in; (void)out_size; (void)ws_size;
    const float* priors   = (const float*)d_in[0];
    // d_in[1] = num_level_priors {6400,1600,400} (compile-time constants here)
    const int*   gtLabels = (const int*)d_in[2];
    const float* gtB      = (const float*)d_in[3];
    const float* pad      = (const float*)d_in[4];
    const float* pred     = (const float*)d_in[5];

    float* dist2   = (float*)d_ws;                       // BG*P floats (then pos_mask)
    float* wsLabel = dist2 + (size_t)BG_ * P_;           // B*P floats
    float* wsIou   = wsLabel + (size_t)B_ * P_;          // B*P floats

    float* out       = (float*)d_out;                    // labels|boxes|scores|fg
    float* outLabels = out;
    float* outBoxes  = out + (size_t)B_ * P_;
    float* outScores = out + (size_t)B_ * P_ * 5;
    float* outFg     = out + (size_t)B_ * P_ * 85;

    dim3 gD(P_ / 16, BG_ / 16);                          // 525 x 64 tiles
    dist2_wmma<<<gD, 32, 0, stream>>>((const float4*)gtB, (const float4*)priors, dist2);

    atss_select<<<BG_, 32, 0, stream>>>(dist2, (const float4*)gtB,
                                        (const float4*)priors, pad);

    const int nT = B_ * P_;
    atss_assign<<<(nT + 255) / 256, 256, 0, stream>>>(
        dist2, (const float4*)gtB, (const float4*)priors, gtLabels,
        (const float4*)pred, outLabels, outBoxes, outFg, wsLabel, wsIou);

    const long nV = (long)nT * (NC_ / 4);
    atss_scores<<<(int)((nV + 255) / 256), 256, 0, stream>>>(
        wsLabel, wsIou, (float4*)outScores);
}